// MaxPooling_7541962572230
// MI455X (gfx1250) — compile-verified
//
#include <hip/hip_runtime.h>
#include <stdint.h>

// ---------------- problem constants (match reference) ----------------
#define NV0   120
#define NV1   160
#define PERB  19200                  // NV0*NV1
#define KC    76800                  // B * NV0 * NV1
#define SENTKEY 5898240000ull        // KC * KC

#define TPB     256
#define RCHUNK  4096                 // keys per block per radix pass
#define RITEMS  16                   // keys per thread  (RCHUNK = TPB*RITEMS)
#define NPASS   5                    // 5 x 8-bit digits covers 40 >= 33 bits

typedef unsigned long long ull;

// ---------------- CDNA5 async-to-LDS availability --------------------
#if defined(__gfx1250__) && defined(__has_builtin)
#if __has_builtin(__builtin_amdgcn_global_load_async_to_lds_b128) && \
    __has_builtin(__builtin_amdgcn_s_wait_asynccnt)
#define HAVE_ASYNC 1
#endif
#endif
#ifndef HAVE_ASYNC
#define HAVE_ASYNC 0
#endif

#if HAVE_ASYNC
// builtin signature (from hipcc diagnostic): param0 = v4i in AS1 (global),
// param1 = v4i in AS3 (LDS), then imm offset, imm cpol.
typedef int v4i __attribute__((ext_vector_type(4)));
typedef __attribute__((address_space(1))) v4i* g_v4i_p;
typedef __attribute__((address_space(3))) v4i* l_v4i_p;
#endif

// ---------------- helpers --------------------------------------------
// order-preserving float<->uint encoding so segment-max can use the
// hardware global_atomic_max_u32 (L2-resident accumulator, 192MB L2)
__device__ __forceinline__ unsigned encF(float f) {
  unsigned u = __float_as_uint(f);
  return (u & 0x80000000u) ? ~u : (u | 0x80000000u);
}
__device__ __forceinline__ float decF(unsigned e) {
  return (e & 0x80000000u) ? __uint_as_float(e ^ 0x80000000u)
                           : __uint_as_float(~e);
}

// ---------------- kernel 0: init accumulators ------------------------
__global__ void k_init(unsigned* __restrict__ xEnc, float* __restrict__ posSum,
                       unsigned* __restrict__ counts, long long nx) {
  long long i = (long long)blockIdx.x * TPB + threadIdx.x;
  if (i >= nx) return;
  xEnc[i] = 0u;                       // encodes "below every real float"
  if (i < (long long)KC * 3) posSum[i] = 0.0f;
  if (i < (long long)KC)     counts[i] = 0u;
}

// ---------------- kernel 1: cluster ids + counts + pos sums ----------
__global__ void k_cluster(const float* __restrict__ pos,
                          const long long* __restrict__ batch,
                          int* __restrict__ cluster,
                          unsigned* __restrict__ counts,
                          float* __restrict__ posSum, long long n) {
  long long i = (long long)blockIdx.x * TPB + threadIdx.x;
  if (i >= n) return;
  float p0 = pos[3 * i + 0];
  float p1 = pos[3 * i + 1];
  float p2 = pos[3 * i + 2];
  int c0 = (int)floorf(p1 * 0.25f);   // pos/4 exact for pow2 divisor
  int c1 = (int)floorf(p2 * 0.25f);
  int b  = (int)batch[i];
  int c  = c0 + c1 * NV0 + b * PERB;
  cluster[i] = c;
  atomicAdd(&counts[c], 1u);                 // global_atomic_add_u32
  atomicAdd(&posSum[3 * c + 0], p0);         // global_atomic_add_f32
  atomicAdd(&posSum[3 * c + 1], p1);
  atomicAdd(&posSum[3 * c + 2], p2);
}

// ---------------- kernel 2: scatter-max of features ------------------
__global__ void k_scatter_max(const float* __restrict__ x,
                              const int* __restrict__ cluster,
                              unsigned* __restrict__ xEnc, long long total) {
  long long i = (long long)blockIdx.x * TPB + threadIdx.x;
  if (i >= total) return;
  long long node = i >> 6;
  int f = (int)(i & 63);
  unsigned c = (unsigned)cluster[node];
  atomicMax(&xEnc[(size_t)c * 64 + f], encF(x[i]));  // global_atomic_max_u32
}

// ---------------- kernel 3: finalize x_out ---------------------------
__global__ void k_final_x(const unsigned* __restrict__ xEnc,
                          const unsigned* __restrict__ counts,
                          float* __restrict__ outX, long long total) {
  long long i = (long long)blockIdx.x * TPB + threadIdx.x;
  if (i >= total) return;
  long long k = i >> 6;
  outX[i] = (counts[k] > 0u) ? decF(xEnc[i]) : 0.0f;
}

// ---------------- kernel 4: finalize pos/batch/mask ------------------
__global__ void k_final_cluster(const float* __restrict__ posSum,
                                const unsigned* __restrict__ counts,
                                float* __restrict__ outPos,
                                float* __restrict__ outBatch,
                                float* __restrict__ outMask) {
  int k = blockIdx.x * TPB + threadIdx.x;
  if (k >= KC) return;
  unsigned c = counts[k];
  float denom = (float)(c > 0u ? c : 1u);
  outPos[3 * k + 0] = posSum[3 * k + 0] / denom;
  outPos[3 * k + 1] = posSum[3 * k + 1] / denom;
  outPos[3 * k + 2] = posSum[3 * k + 2] / denom;
  outBatch[k] = (float)(k / PERB);
  outMask[k]  = (c > 0u) ? 1.0f : 0.0f;
}

// ---------------- kernel 5: build edge sort keys ---------------------
__global__ void k_edge_keys(const long long* __restrict__ eidx,
                            const int* __restrict__ cluster,
                            ull* __restrict__ keys, long long E) {
  long long e = (long long)blockIdx.x * TPB + threadIdx.x;
  if (e >= E) return;
  long long s = eidx[e];
  long long d = eidx[E + e];
  unsigned sc = (unsigned)cluster[s];
  unsigned dc = (unsigned)cluster[d];
  keys[e] = (sc != dc) ? ((ull)sc * (ull)KC + (ull)dc) : SENTKEY;
}

// ---------------- radix pass kernels ---------------------------------
__global__ void k_radix_hist(const ull* __restrict__ src,
                             unsigned* __restrict__ hist, long long n, int nb,
                             int shift) {
  __shared__ unsigned h[256];
  int t = threadIdx.x;
  h[t] = 0u;
  __syncthreads();
  long long base = (long long)blockIdx.x * RCHUNK;
  for (int i = 0; i < RITEMS; ++i) {
    long long gi = base + t + i * TPB;
    if (gi < n) {
      unsigned d = (unsigned)(src[gi] >> shift) & 255u;
      atomicAdd(&h[d], 1u);                          // ds_add_u32
    }
  }
  __syncthreads();
  hist[(size_t)t * nb + blockIdx.x] = h[t];          // digit-major layout
}

// exclusive scan within each digit's row across blocks; one block/digit
__global__ void k_scan_rows(const unsigned* __restrict__ hist,
                            unsigned* __restrict__ within,
                            unsigned* __restrict__ digitTotal, int nb) {
  __shared__ unsigned sh[256];
  int d = blockIdx.x;
  int t = threadIdx.x;
  unsigned carry = 0u;
  for (int c0 = 0; c0 < nb; c0 += 256) {
    int i = c0 + t;
    unsigned v = (i < nb) ? hist[(size_t)d * nb + i] : 0u;
    sh[t] = v;
    __syncthreads();
    for (int off = 1; off < 256; off <<= 1) {
      unsigned u = (t >= off) ? sh[t - off] : 0u;
      __syncthreads();
      sh[t] += u;
      __syncthreads();
    }
    unsigned incl  = sh[t];
    unsigned total = sh[255];
    if (i < nb) within[(size_t)d * nb + i] = carry + incl - v;
    carry += total;
    __syncthreads();
  }
  if (t == 0) digitTotal[d] = carry;
}

__global__ void k_scan_digits(const unsigned* __restrict__ digitTotal,
                              unsigned* __restrict__ digitBase) {
  __shared__ unsigned sh[256];
  int t = threadIdx.x;
  unsigned v = digitTotal[t];
  sh[t] = v;
  __syncthreads();
  for (int off = 1; off < 256; off <<= 1) {
    unsigned u = (t >= off) ? sh[t - off] : 0u;
    __syncthreads();
    sh[t] += u;
    __syncthreads();
  }
  digitBase[t] = sh[t] - v;                          // exclusive
}

// stable scatter: wave32 ballot ranking; wave segments keep global order.
__global__ void k_radix_scatter(const ull* __restrict__ src,
                                ull* __restrict__ dst,
                                const unsigned* __restrict__ within,
                                const unsigned* __restrict__ digitBase,
                                long long n, int nb, int shift) {
  __shared__ __attribute__((aligned(16))) ull skeys[RCHUNK];
  __shared__ unsigned waveHist[8][256];
  __shared__ unsigned blockBase[256];

  const int t = threadIdx.x;
  const int lane = t & 31;
  const int w = t >> 5;
  const long long blockStart = (long long)blockIdx.x * RCHUNK;
  const long long segStart = blockStart + (long long)w * 512;

  for (int i = 0; i < 8; ++i) waveHist[i][t] = 0u;

  bool fullBlock = (blockStart + RCHUNK) <= n;
#if HAVE_ASYNC
  if (fullBlock) {
    // CDNA5 async DMA: stage this wave's 4KB key segment into LDS.
    char* g = const_cast<char*>((const char*)(src + segStart));
    char* l = (char*)&skeys[(size_t)w * 512];
    for (int i = 0; i < 8; ++i) {
      int bo = (lane + i * 32) * 16;
      g_v4i_p gp = (g_v4i_p)(g + bo);
      l_v4i_p lp = (l_v4i_p)(l + bo);
      __builtin_amdgcn_global_load_async_to_lds_b128(gp, lp, 0, 0);
    }
  } else {
    for (int i = 0; i < RITEMS; ++i) {
      int li = w * 512 + lane + i * 32;
      long long gi = blockStart + li;
      skeys[li] = (gi < n) ? src[gi] : ~0ull;
    }
  }
#else
  for (int i = 0; i < RITEMS; ++i) {
    int li = w * 512 + lane + i * 32;
    long long gi = blockStart + li;
    skeys[li] = (gi < n) ? src[gi] : ~0ull;
  }
#endif
  __syncthreads();
#if HAVE_ASYNC
  __builtin_amdgcn_s_wait_asynccnt(0);
  asm volatile("" ::: "memory");
#endif

  unsigned meta[RITEMS];
  const unsigned lowMask = (1u << lane) - 1u;
  for (int r = 0; r < RITEMS; ++r) {
    int li = w * 512 + r * 32 + lane;
    long long gi = blockStart + li;
    bool valid = gi < n;
    ull key = skeys[li];
    unsigned digit = (unsigned)(key >> shift) & 255u;
    // match lanes sharing this 8-bit digit (wave32 ballots)
    unsigned mask = (unsigned)__ballot(valid);
    for (int b = 0; b < 8; ++b) {
      unsigned bb = (unsigned)__ballot((digit >> b) & 1u);
      mask &= ((digit >> b) & 1u) ? bb : ~bb;
    }
    unsigned rankInRound = __popc(mask & lowMask);
    int leader = __ffs(mask) - 1;
    unsigned cnt = __popc(mask);
    unsigned prev = 0u;
    if (valid && lane == leader)
      prev = atomicAdd(&waveHist[w][digit], cnt);    // ds_add_rtn_u32
    prev = (unsigned)__shfl((int)prev, leader, 32);
    meta[r] = valid ? ((digit << 16) | (prev + rankInRound)) : 0xFFFFFFFFu;
  }
  __syncthreads();

  // per-digit: exclusive prefix across waves + global base for this block
  {
    unsigned run = 0u;
    for (int ww = 0; ww < 8; ++ww) {
      unsigned v = waveHist[ww][t];
      waveHist[ww][t] = run;
      run += v;
    }
    blockBase[t] = digitBase[t] + within[(size_t)t * nb + blockIdx.x];
  }
  __syncthreads();

  for (int r = 0; r < RITEMS; ++r) {
    unsigned m = meta[r];
    if (m == 0xFFFFFFFFu) continue;
    int li = w * 512 + r * 32 + lane;
    unsigned digit = m >> 16;
    unsigned pos = blockBase[digit] + waveHist[w][digit] + (m & 0xFFFFu);
    dst[pos] = skeys[li];
  }
}

// ---------------- kernel: emit ei / edge_mask from sorted keys -------
__global__ void k_edge_out(const ull* __restrict__ sorted,
                           float* __restrict__ outEi,
                           float* __restrict__ outMask, long long E) {
  long long e = (long long)blockIdx.x * TPB + threadIdx.x;
  if (e >= E) return;
  ull key = sorted[e];
  bool first = (e == 0) || (key != sorted[e - 1]);
  bool m = first && (key < SENTKEY);
  float a = -1.0f, b = -1.0f;
  if (m) {
    unsigned q = (unsigned)(key / (ull)KC);
    unsigned r = (unsigned)(key % (ull)KC);
    a = (float)q;                      // < 2^24 -> exact in f32
    b = (float)r;
  }
  outEi[e]     = a;
  outEi[E + e] = b;
  outMask[e]   = m ? 1.0f : 0.0f;
}

// ---------------- launcher -------------------------------------------
extern "C" void kernel_launch(void* const* d_in, const int* in_sizes, int n_in,
                              void* d_out, int out_size, void* d_ws,
                              size_t ws_size, hipStream_t stream) {
  const float* x          = (const float*)d_in[0];
  const float* pos        = (const float*)d_in[1];
  const long long* batch  = (const long long*)d_in[2];
  const long long* eidx   = (const long long*)d_in[3];

  const long long N = (long long)in_sizes[0] / 64;
  const long long E = (long long)in_sizes[3] / 2;
  const int NB = (int)((E + RCHUNK - 1) / RCHUNK);

  // ---- workspace carve (256B aligned) ----
  char* w = (char*)d_ws;
  auto sub = [&](size_t bytes) -> char* {
    char* p = w;
    w += (bytes + 255) & ~(size_t)255;
    return p;
  };
  int*      cluster    = (int*)sub((size_t)N * 4);
  unsigned* counts     = (unsigned*)sub((size_t)KC * 4);
  float*    posSum     = (float*)sub((size_t)KC * 12);
  unsigned* xEnc       = (unsigned*)sub((size_t)KC * 64 * 4);
  unsigned* hist       = (unsigned*)sub((size_t)256 * NB * 4);
  unsigned* within     = (unsigned*)sub((size_t)256 * NB * 4);
  unsigned* digitTotal = (unsigned*)sub(1024);
  unsigned* digitBase  = (unsigned*)sub(1024);
  ull*      keysA      = (ull*)sub((size_t)E * 8);
  ull*      keysB      = (ull*)sub((size_t)E * 8);

  // ---- output layout (flat float, return order) ----
  float* out      = (float*)d_out;
  float* outX     = out;                                  // K*64
  float* outPos   = out + (size_t)KC * 64;                // K*3
  float* outBatch = outPos + (size_t)KC * 3;              // K
  float* outNMask = outBatch + (size_t)KC;                // K
  float* outEi    = outNMask + (size_t)KC;                // 2*E
  float* outEMask = outEi + (size_t)2 * E;                // E

  const long long nx = (long long)KC * 64;
  int gInit  = (int)((nx + TPB - 1) / TPB);
  int gNode  = (int)((N + TPB - 1) / TPB);
  int gXF    = (int)((N * 64 + TPB - 1) / TPB);
  int gK     = (KC + TPB - 1) / TPB;
  int gE     = (int)((E + TPB - 1) / TPB);

  // pooled-node pipeline
  k_init<<<gInit, TPB, 0, stream>>>(xEnc, posSum, counts, nx);
  k_cluster<<<gNode, TPB, 0, stream>>>(pos, batch, cluster, counts, posSum, N);
  k_scatter_max<<<gXF, TPB, 0, stream>>>(x, cluster, xEnc, N * 64);
  k_final_x<<<gInit, TPB, 0, stream>>>(xEnc, counts, outX, nx);
  k_final_cluster<<<gK, TPB, 0, stream>>>(posSum, counts, outPos, outBatch,
                                          outNMask);

  // edge pipeline: keys -> 5-pass LSD radix (stable) -> dedupe/emit
  k_edge_keys<<<gE, TPB, 0, stream>>>(eidx, cluster, keysA, E);
  ull* cur = keysA;
  ull* nxt = keysB;
  for (int p = 0; p < NPASS; ++p) {
    int shift = 8 * p;
    k_radix_hist<<<NB, TPB, 0, stream>>>(cur, hist, E, NB, shift);
    k_scan_rows<<<256, TPB, 0, stream>>>(hist, within, digitTotal, NB);
    k_scan_digits<<<1, TPB, 0, stream>>>(digitTotal, digitBase);
    k_radix_scatter<<<NB, TPB, 0, stream>>>(cur, nxt, within, digitBase, E, NB,
                                            shift);
    ull* t2 = cur;
    cur = nxt;
    nxt = t2;
  }
  k_edge_out<<<gE, TPB, 0, stream>>>(cur, outEi, outEMask, E);

  (void)n_in;
  (void)out_size;
  (void)ws_size;
}